// Attention_59296318489080
// MI455X (gfx1250) — compile-verified
//
#include <hip/hip_runtime.h>
#include <hip/hip_bf16.h>
#include <math.h>

// ---------------------------------------------------------------------------
// Problem constants (from reference): B=32, SQ=1, HID=4096, HQ=32, HKV=8,
// DH=128, PAST=4096, T=4097, SCALE=DH^-0.5
// ---------------------------------------------------------------------------
#define B_    32
#define HID_  4096
#define HQ_   32
#define HKV_  8
#define DH_   128
#define PAST_ 4096
#define T_    4097
#define SCALE_ 0.08838834764831845f   // 128^-0.5

// d_out layout (floats): [attn_output 32*4096][present_key 256*4097*128][present_value ...]
#define OUT_ATTN 0LL
#define OUT_PK   131072LL
#define OUT_PV   (131072LL + 256LL * 4097LL * 128LL)   // 134381568

// workspace layout (floats)
#define WS_QLIN 0LL
#define WS_KLIN 131072LL
#define WS_VLIN 163840LL
#define WS_QROT 196608LL
#define WS_CTX  327680LL

typedef __attribute__((ext_vector_type(2))) float v2f;
typedef __attribute__((ext_vector_type(4))) float v4f;
typedef __attribute__((ext_vector_type(8))) float v8f;

// ---------------------------------------------------------------------------
// GEMM: C[M,N] = X[M,K] * W[N,K]^T   (both operands K-contiguous -> WMMA NT)
// fp32 WMMA 16x16x4. One 16x16 output tile per wave, 8 waves per block.
// A frag (16x4): lane l holds A[l&15, (l>>4)*2 + {0,1}] -> one float2 load.
// B frag (4x16): lane l holds B[(l>>4)*2 + {0,1}, l&15] = W[n, k...] -> float2.
// C/D (16x16 f32): VGPR j: lanes0-15 row j, lanes16-31 row j+8.
// ---------------------------------------------------------------------------
__global__ __launch_bounds__(256) void gemm_xwt_wmma(
    const float* __restrict__ X, const float* __restrict__ W,
    float* __restrict__ C, int M, int N, int K) {
  const int wave = blockIdx.x * (blockDim.x >> 5) + (threadIdx.x >> 5);
  const int lane = threadIdx.x & 31;
  const int ntiles_n = N >> 4;
  const int total = (M >> 4) * ntiles_n;
  if (wave >= total) return;                 // wave-uniform: EXEC stays all-ones
  const int mt = wave / ntiles_n;
  const int nt = wave - mt * ntiles_n;

  const int khalf = (lane >> 4) << 1;        // 0 or 2
  const float* __restrict__ xp = X + (size_t)(mt * 16 + (lane & 15)) * K + khalf;
  const float* __restrict__ wp = W + (size_t)(nt * 16 + (lane & 15)) * K + khalf;

  v8f c0 = {};
  v8f c1 = {};
#pragma unroll 4
  for (int k0 = 0; k0 < K; k0 += 8) {
    v2f a0 = *(const v2f*)(xp + k0);
    v2f b0 = *(const v2f*)(wp + k0);
    v2f a1 = *(const v2f*)(xp + k0 + 4);
    v2f b1 = *(const v2f*)(wp + k0 + 4);
    c0 = __builtin_amdgcn_wmma_f32_16x16x4_f32(false, a0, false, b0,
                                               (short)0, c0, false, false);
    c1 = __builtin_amdgcn_wmma_f32_16x16x4_f32(false, a1, false, b1,
                                               (short)0, c1, false, false);
  }
  c0 += c1;

  const int crow0 = mt * 16 + ((lane >> 4) << 3);
  const int ccol  = nt * 16 + (lane & 15);
#pragma unroll
  for (int j = 0; j < 8; ++j)
    C[(size_t)(crow0 + j) * N + ccol] = c0[j];
}

// ---------------------------------------------------------------------------
// RoPE on Q: qrot[b,h,d] from qlin[b, h*128+d]. One thread per (b,h,d<64).
// ---------------------------------------------------------------------------
__global__ void rope_q_kernel(const float* __restrict__ qlin,
                              float* __restrict__ qrot,
                              const int* __restrict__ pos,
                              const float* __restrict__ cosc,
                              const float* __restrict__ sinc) {
  int idx = blockIdx.x * blockDim.x + threadIdx.x;   // B*HQ*64 = 65536
  if (idx >= B_ * HQ_ * 64) return;
  int d = idx & 63;
  int h = (idx >> 6) & 31;
  int b = idx >> 11;
  int p = pos[b];
  float cs = cosc[(size_t)p * 64 + d];
  float sn = sinc[(size_t)p * 64 + d];
  const float* src = qlin + ((size_t)b * HQ_ + h) * DH_;
  float x1 = src[d], x2 = src[d + 64];
  float* dst = qrot + ((size_t)b * HQ_ + h) * DH_;
  dst[d]      = x1 * cs - x2 * sn;
  dst[d + 64] = x2 * cs + x1 * sn;
}

// ---------------------------------------------------------------------------
// RoPE on new K + passthrough V, written straight into present_* at t=PAST.
// One thread per (b,h,d<64).
// ---------------------------------------------------------------------------
__global__ void rope_kv_kernel(const float* __restrict__ klin,
                               const float* __restrict__ vlin,
                               float* __restrict__ out,
                               const int* __restrict__ pos,
                               const float* __restrict__ cosc,
                               const float* __restrict__ sinc) {
  int idx = blockIdx.x * blockDim.x + threadIdx.x;   // B*HKV*64 = 16384
  if (idx >= B_ * HKV_ * 64) return;
  int d = idx & 63;
  int h = (idx >> 6) & 7;
  int b = idx >> 9;
  int p = pos[b];
  float cs = cosc[(size_t)p * 64 + d];
  float sn = sinc[(size_t)p * 64 + d];
  const float* ks = klin + ((size_t)b * HKV_ + h) * DH_;
  float x1 = ks[d], x2 = ks[d + 64];
  size_t slot = (((size_t)(b * HKV_ + h)) * T_ + PAST_) * DH_;
  float* pk = out + OUT_PK + slot;
  pk[d]      = x1 * cs - x2 * sn;
  pk[d + 64] = x2 * cs + x1 * sn;
  const float* vs = vlin + ((size_t)b * HKV_ + h) * DH_;
  float* pv = out + OUT_PV + slot;
  pv[d]      = vs[d];
  pv[d + 64] = vs[d + 64];
}

// ---------------------------------------------------------------------------
// Fused flash-decode + KV-cache copy-out.
// Block = (b, hkv); 256 threads = 8 waves; waves split T contiguously.
// Lane owns DH slice [4*lane, 4*lane+4): coalesced b128 streaming of K/V with
// non-temporal hints (2 GB one-pass stream; keep it out of L2), past rows
// stored to present_* in the same pass (single HBM read total).
// Per key: 4 grouped-head dot4s -> butterfly wave reduction -> online softmax.
// Cross-wave combine via LDS. Only wave 7 touches the new token (t=4096),
// handled as a tail step so the hot loop is branch-free.
// ---------------------------------------------------------------------------
__global__ __launch_bounds__(256) void attn_decode_kernel(
    const float* __restrict__ past_key, const float* __restrict__ past_value,
    const float* __restrict__ bias, const float* __restrict__ qrot,
    float* __restrict__ out, float* __restrict__ ctx) {
  const int bh = blockIdx.x;                 // 0..255
  const int b = bh >> 3, h = bh & 7;
  const int w = threadIdx.x >> 5, lane = threadIdx.x & 31;

  // q for 4 grouped heads, pre-scaled
  float q[4][4];
#pragma unroll
  for (int g = 0; g < 4; ++g) {
    const float* qp = qrot + ((size_t)b * HQ_ + h * 4 + g) * DH_ + lane * 4;
#pragma unroll
    for (int j = 0; j < 4; ++j) q[g][j] = qp[j] * SCALE_;
  }

  const float* kpast = past_key   + (size_t)bh * PAST_ * DH_;
  const float* vpast = past_value + (size_t)bh * PAST_ * DH_;
  float* pk_out = out + OUT_PK + (size_t)bh * T_ * DH_;
  float* pv_out = out + OUT_PV + (size_t)bh * T_ * DH_;
  const float* brow = bias + (size_t)b * T_;

  const int chunk = (T_ + 7) / 8;            // 513
  const int t0 = w * chunk;
  const int t1 = (t0 + chunk < T_) ? (t0 + chunk) : T_;
  const int tmain = (t1 < PAST_) ? t1 : PAST_;

  float m = -INFINITY, l = 0.0f;
  float acc[4][4] = {};

  auto body = [&](v4f kv, v4f vv, int t) {
    float s[4];
#pragma unroll
    for (int g = 0; g < 4; ++g)
      s[g] = kv.x * q[g][0] + kv.y * q[g][1] + kv.z * q[g][2] + kv.w * q[g][3];
#pragma unroll
    for (int off = 16; off > 0; off >>= 1)
#pragma unroll
      for (int g = 0; g < 4; ++g)
        s[g] += __shfl_xor(s[g], off, 32);

    const float bb = brow[t];
    float smax = -INFINITY;
#pragma unroll
    for (int g = 0; g < 4; ++g) { s[g] += bb; smax = fmaxf(smax, s[g]); }

    const float nm = fmaxf(m, smax);
    const float corr = __expf(m - nm);
    float p[4];
    l *= corr;
#pragma unroll
    for (int g = 0; g < 4; ++g) { p[g] = __expf(s[g] - nm); l += p[g]; }
    const float va[4] = {vv.x, vv.y, vv.z, vv.w};
#pragma unroll
    for (int g = 0; g < 4; ++g)
#pragma unroll
      for (int j = 0; j < 4; ++j)
        acc[g][j] = acc[g][j] * corr + p[g] * va[j];
    m = nm;
  };

#pragma unroll 2
  for (int t = t0; t < tmain; ++t) {
    const size_t off = (size_t)t * DH_ + lane * 4;
    v4f kv = __builtin_nontemporal_load((const v4f*)(kpast + off));
    v4f vv = __builtin_nontemporal_load((const v4f*)(vpast + off));
    __builtin_nontemporal_store(kv, (v4f*)(pk_out + off));   // fused copy-out
    __builtin_nontemporal_store(vv, (v4f*)(pv_out + off));
    body(kv, vv, t);
  }

  if (t1 == T_) {                            // new token, written by rope_kv
    const size_t off = (size_t)PAST_ * DH_ + lane * 4;
    v4f kv = *(const v4f*)(pk_out + off);
    v4f vv = *(const v4f*)(pv_out + off);
    body(kv, vv, PAST_);
  }

  // cross-wave combine
  __shared__ float sm[8], sl[8];
  __shared__ float sacc[8 * 512];
  if (lane == 0) { sm[w] = m; sl[w] = l; }
#pragma unroll
  for (int g = 0; g < 4; ++g)
#pragma unroll
    for (int j = 0; j < 4; ++j)
      sacc[w * 512 + g * 128 + lane * 4 + j] = acc[g][j];
  __syncthreads();

  float gm = -INFINITY;
#pragma unroll
  for (int w2 = 0; w2 < 8; ++w2) gm = fmaxf(gm, sm[w2]);
  float f[8];
  float L = 0.0f;
#pragma unroll
  for (int w2 = 0; w2 < 8; ++w2) { f[w2] = __expf(sm[w2] - gm); L += sl[w2] * f[w2]; }
  const float invL = 1.0f / L;

  for (int e = threadIdx.x; e < 512; e += 256) {
    float v = 0.0f;
#pragma unroll
    for (int w2 = 0; w2 < 8; ++w2) v += sacc[w2 * 512 + e] * f[w2];
    ctx[(size_t)b * (HQ_ * DH_) + h * 512 + e] = v * invL;
  }
}

// ---------------------------------------------------------------------------
extern "C" void kernel_launch(void* const* d_in, const int* in_sizes, int n_in,
                              void* d_out, int out_size, void* d_ws, size_t ws_size,
                              hipStream_t stream) {
  const float* hidden = (const float*)d_in[0];   // [32,1,4096]
  const float* bias   = (const float*)d_in[1];   // [32,1,1,4097]
  const int*   pos    = (const int*)d_in[2];     // [32,1]
  const float* cosc   = (const float*)d_in[3];   // [8192,64]
  const float* sinc   = (const float*)d_in[4];   // [8192,64]
  const float* pastk  = (const float*)d_in[5];   // [32,8,4096,128]
  const float* pastv  = (const float*)d_in[6];   // [32,8,4096,128]
  const float* Wq     = (const float*)d_in[7];   // [4096,4096]
  const float* Wk     = (const float*)d_in[8];   // [1024,4096]
  const float* Wv     = (const float*)d_in[9];   // [1024,4096]
  const float* Wo     = (const float*)d_in[10];  // [4096,4096]
  float* out = (float*)d_out;
  float* ws  = (float*)d_ws;

  float* qlin = ws + WS_QLIN;
  float* klin = ws + WS_KLIN;
  float* vlin = ws + WS_VLIN;
  float* qrot = ws + WS_QROT;
  float* ctx  = ws + WS_CTX;

  // 1-3: QKV projections (WMMA fp32). tiles = (M/16)*(N/16); 8 waves/block.
  {
    int tiles_q = (B_ / 16) * (HID_ / 16);          // 512
    int tiles_kv = (B_ / 16) * ((HKV_ * DH_) / 16); // 128
    gemm_xwt_wmma<<<(tiles_q + 7) / 8, 256, 0, stream>>>(hidden, Wq, qlin, B_, HQ_ * DH_, HID_);
    gemm_xwt_wmma<<<(tiles_kv + 7) / 8, 256, 0, stream>>>(hidden, Wk, klin, B_, HKV_ * DH_, HID_);
    gemm_xwt_wmma<<<(tiles_kv + 7) / 8, 256, 0, stream>>>(hidden, Wv, vlin, B_, HKV_ * DH_, HID_);
  }

  // 4-5: RoPE; new K/V written into present_* slot t=4096
  rope_q_kernel<<<(B_ * HQ_ * 64) / 256, 256, 0, stream>>>(qlin, qrot, pos, cosc, sinc);
  rope_kv_kernel<<<(B_ * HKV_ * 64) / 256, 256, 0, stream>>>(klin, vlin, out, pos, cosc, sinc);

  // 6: fused flash-decode + KV-cache copy-out
  attn_decode_kernel<<<B_ * HKV_, 256, 0, stream>>>(pastk, pastv, bias, qrot, out, ctx);

  // 7: output projection into d_out[0 : 32*4096]
  {
    int tiles_o = (B_ / 16) * (HID_ / 16);          // 512
    gemm_xwt_wmma<<<(tiles_o + 7) / 8, 256, 0, stream>>>(ctx, Wo, out + OUT_ATTN, B_, HID_, HID_);
  }
}